// Sender_Receiver_35055523070769
// MI455X (gfx1250) — compile-verified
//
#include <hip/hip_runtime.h>

typedef __attribute__((ext_vector_type(2))) float v2f;
typedef __attribute__((ext_vector_type(8))) float v8f;

constexpr int kH   = 100;   // hidden
constexpr int kV   = 20;    // vocab
constexpr int kG   = 300;   // 3*H (gates r,z,n stacked)
constexpr int kGPad = 19 * 16;      // 304, padded gate buffer
constexpr int kThreads = 320;       // 10 wave32

__device__ __forceinline__ int iminc(int a, int b) { return a < b ? a : b; }

// ---------------------------------------------------------------------------
// out[m0 .. m0+31] = W[nrows x 100] (row-major) @ hv[100], two 16-row tiles
// per wave, via V_WMMA_F32_16X16X4_F32 (full fp32 — no precision loss).
//   A layout (ISA 7.12.2 16x4 f32): lane L, vgpr v -> M = L%16, K = 2*(L/16)+v
//   B layout (symmetric):           lane L, vgpr v -> N = L%16, K = 2*(L/16)+v
//     (h broadcast across all N columns, so every D column = the matvec)
//   D layout: lane L, vgpr v -> M = v + 8*(L/16), N = L%16
// Out-of-range rows are CLAMPED (not zeroed): their D rows are garbage but the
// writeback guard skips them, so A loads are unconditional -> no EXEC juggling.
// ---------------------------------------------------------------------------
__device__ __forceinline__ void wmma_two_tile_matvec100(
    const float* __restrict__ W, int nrows, const float* hv,
    float* outv, int m0, int lane)
{
  const int khalf = (lane >> 4) << 1;                     // 0 or 2
  const int mr0 = iminc(m0 +      (lane & 15), nrows - 1);
  const int mr1 = iminc(m0 + 16 + (lane & 15), nrows - 1);
  const float* r0 = W + (long)mr0 * 100 + khalf;          // 8B-aligned
  const float* r1 = W + (long)mr1 * 100 + khalf;
  const float* hb = hv + khalf;

  v8f acc0 = {0.f, 0.f, 0.f, 0.f, 0.f, 0.f, 0.f, 0.f};
  v8f acc1 = {0.f, 0.f, 0.f, 0.f, 0.f, 0.f, 0.f, 0.f};
  #pragma unroll
  for (int k0 = 0; k0 < 100; k0 += 4) {
    const v2f b  = *(const v2f*)(hb + k0);                // ds_load_b64 (shared)
    const v2f a0 = *(const v2f*)(r0 + k0);                // global_load_b64
    const v2f a1 = *(const v2f*)(r1 + k0);
    acc0 = __builtin_amdgcn_wmma_f32_16x16x4_f32(false, a0, false, b,
                                                 (short)0, acc0, false, false);
    acc1 = __builtin_amdgcn_wmma_f32_16x16x4_f32(false, a1, false, b,
                                                 (short)0, acc1, false, false);
  }
  // Column N==0 lanes write out: lane 0 -> M=+0..7, lane 16 -> M=+8..15.
  if ((lane & 15) == 0) {
    const int mb = (lane >> 4) << 3;
    #pragma unroll
    for (int v = 0; v < 8; ++v) {
      const int ma = m0 + mb + v;
      const int mc = m0 + 16 + mb + v;
      if (ma < nrows) outv[ma] = acc0[v];
      if (mc < nrows) outv[mc] = acc1[v];
    }
  }
}

// Single-tile variant for the 20-row logits matvec (waves 0-1).
__device__ __forceinline__ void wmma_tile_matvec100(
    const float* __restrict__ W, int nrows, const float* hv,
    float* outv, int m0, int lane)
{
  const int khalf = (lane >> 4) << 1;
  const int mr = iminc(m0 + (lane & 15), nrows - 1);
  const float* r0 = W + (long)mr * 100 + khalf;
  const float* hb = hv + khalf;

  v8f acc = {0.f, 0.f, 0.f, 0.f, 0.f, 0.f, 0.f, 0.f};
  #pragma unroll
  for (int k0 = 0; k0 < 100; k0 += 4) {
    const v2f b = *(const v2f*)(hb + k0);
    const v2f a = *(const v2f*)(r0 + k0);
    acc = __builtin_amdgcn_wmma_f32_16x16x4_f32(false, a, false, b,
                                                (short)0, acc, false, false);
  }
  if ((lane & 15) == 0) {
    const int mb = (lane >> 4) << 3;
    #pragma unroll
    for (int v = 0; v < 8; ++v) {
      const int m = m0 + mb + v;
      if (m < nrows) outv[m] = acc[v];
    }
  }
}

// PyTorch GRUCell gate math. gi already includes bih; gh = whh@h (bhh added here).
__device__ __forceinline__ void gru_update(int t, const float* gi, const float* gh,
                                           const float* __restrict__ bhh, float* h)
{
  const float hr = gh[t]          + bhh[t];
  const float hz = gh[kH + t]     + bhh[kH + t];
  const float hn = gh[2 * kH + t] + bhh[2 * kH + t];
  const float r  = 1.f / (1.f + expf(-(gi[t] + hr)));
  const float z  = 1.f / (1.f + expf(-(gi[kH + t] + hz)));
  const float n  = tanhf(gi[2 * kH + t] + r * hn);
  h[t] = (1.f - z) * n + z * h[t];
}

__global__ __launch_bounds__(kThreads) void Sender_Receiver_35055523070769_kernel(
    const float* __restrict__ x,     const float* __restrict__ gumbel,
    const float* __restrict__ w1,    const float* __restrict__ b1,
    const float* __restrict__ s_wih, const float* __restrict__ s_whh,
    const float* __restrict__ s_bih, const float* __restrict__ s_bhh,
    const float* __restrict__ w2,    const float* __restrict__ b2,
    const float* __restrict__ r_wih, const float* __restrict__ r_whh,
    const float* __restrict__ r_bih, const float* __restrict__ r_bhh,
    const float* __restrict__ w3,    const float* __restrict__ b3,
    float* __restrict__ out)
{
  __shared__ float sh[kH];            // sender hidden
  __shared__ float rh[kH];            // receiver hidden
  __shared__ float gh[kGPad];         // whh@h result (padded to 304)
  __shared__ float gi[kG];            // wih-side gates (bih folded in)
  __shared__ float logits[32];        // w2@sh (padded 20 -> 32)
  __shared__ int   s_idx;
  __shared__ int   s_sym[2];

  const int t    = threadIdx.x;
  const int lane = t & 31;
  const int wave = t >> 5;
  const int m0   = wave * 32;         // each wave owns row-tiles m0, m0+16

  if (t < kH) {
    sh[t] = x[0] * w1[t] + b1[t];     // sender_h = x @ w1.T + b1
    rh[t] = 0.f;
  }
  if (t == 0) s_idx = -1;             // symbol starts as all-zeros
  __syncthreads();

  #pragma unroll 1
  for (int step = 0; step < 2; ++step) {
    // ---- Phase 1: sender input gates (one-hot gather) + gh = s_whh @ sh
    {
      const int idx = s_idx;
      if (t < kG) {
        float g = s_bih[t];
        if (idx >= 0) g += s_wih[t * kV + idx];   // symbol is exactly one-hot
        gi[t] = g;
      }
      wmma_two_tile_matvec100(s_whh, kG, sh, gh, m0, lane);  // branch-free
    }
    __syncthreads();

    // ---- Phase 2: sender GRU update
    if (t < kH) gru_update(t, gi, gh, s_bhh, sh);
    __syncthreads();

    // ---- Phase 3: logits = w2 @ sh (20 rows -> 2 tiles, waves 0-1)
    if (wave < 2) wmma_tile_matvec100(w2, kV, sh, logits, wave * 16, lane);
    __syncthreads();

    // ---- Phase 4: argmax(logits + b2 + gumbel) == gumbel-softmax-hard index
    if (t == 0) {
      const float* g = gumbel + step * kV;
      int   best = 0;
      float bv   = logits[0] + b2[0] + g[0];
      for (int j = 1; j < kV; ++j) {
        const float v = logits[j] + b2[j] + g[j];
        if (v > bv) { bv = v; best = j; }
      }
      s_idx = best;
      s_sym[step] = best;
    }
    __syncthreads();

    // ---- Phase 5: receiver input gates + gh = r_whh @ rh
    {
      const int idx = s_idx;
      if (t < kG) gi[t] = r_bih[t] + r_wih[t * kV + idx];
      wmma_two_tile_matvec100(r_whh, kG, rh, gh, m0, lane);
    }
    __syncthreads();

    // ---- Phase 6: receiver GRU update
    if (t < kH) gru_update(t, gi, gh, r_bhh, rh);
    __syncthreads();
  }

  // prediction = receiver_h @ w3.T + b3 ; outputs: [pred, sym0, sym1]
  if (t == 0) {
    float acc = b3[0];
    for (int k = 0; k < kH; ++k) acc += w3[k] * rh[k];
    out[0] = acc;
    out[1] = (float)s_sym[0];
    out[2] = (float)s_sym[1];
  }
}

extern "C" void kernel_launch(void* const* d_in, const int* in_sizes, int n_in,
                              void* d_out, int out_size, void* d_ws, size_t ws_size,
                              hipStream_t stream) {
  (void)in_sizes; (void)n_in; (void)out_size; (void)d_ws; (void)ws_size;
  Sender_Receiver_35055523070769_kernel<<<1, kThreads, 0, stream>>>(
      (const float*)d_in[0],  (const float*)d_in[1],  (const float*)d_in[2],
      (const float*)d_in[3],  (const float*)d_in[4],  (const float*)d_in[5],
      (const float*)d_in[6],  (const float*)d_in[7],  (const float*)d_in[8],
      (const float*)d_in[9],  (const float*)d_in[10], (const float*)d_in[11],
      (const float*)d_in[12], (const float*)d_in[13], (const float*)d_in[14],
      (const float*)d_in[15], (float*)d_out);
}